// SMPLModel_24850680774801
// MI455X (gfx1250) — compile-verified
//
#include <hip/hip_runtime.h>
#include <math.h>

typedef float v2f __attribute__((ext_vector_type(2)));
typedef float v8f __attribute__((ext_vector_type(8)));

#define V_N 50000
#define K_J 24
#define NB_ 10
#define P_  207   // (K-1)*9

// workspace layout (float offsets)
#define OFF_VSH  0                   // V_N*3 = 150000 floats
#define OFF_J    (V_N * 3)           // 72 floats (24 joints x 3)
#define OFF_LROT (OFF_J + 80)        // 207 floats
#define OFF_RES  (OFF_LROT + 208)    // 24*16 = 384 floats

// ---------------------------------------------------------------- kernel 0
__global__ void k_zeroJ(float* __restrict__ ws) {
    int t = threadIdx.x;
    if (t < 72) ws[OFF_J + t] = 0.0f;
}

// ---------------------------------------------------------------- kernel 1
// v_shaped = einsum('vki,i->vk', shapedirs, betas) + v_template
// J       += J_regressor @ v_shaped   (block-reduced, then global atomics)
__global__ void k_vshaped_J(const float* __restrict__ betas,
                            const float* __restrict__ shapedirs,
                            const float* __restrict__ v_template,
                            const float* __restrict__ Jreg,
                            float* __restrict__ ws) {
    __shared__ float sb[NB_];
    __shared__ float jacc[72];
    int t = threadIdx.x;
    if (t < NB_) sb[t] = betas[t];
    if (t < 72)  jacc[t] = 0.0f;
    __syncthreads();

    int v   = blockIdx.x * blockDim.x + t;
    bool act = (v < V_N);
    int  vc  = act ? v : (V_N - 1);

    float vs[3];
    #pragma unroll
    for (int c = 0; c < 3; ++c) {
        const float* sd = shapedirs + (size_t)(vc * 3 + c) * NB_;
        float a = v_template[vc * 3 + c];
        #pragma unroll
        for (int i = 0; i < NB_; ++i) a += sd[i] * sb[i];
        vs[c] = a;
        if (act) ws[OFF_VSH + v * 3 + c] = a;
    }

    if (act) {
        for (int j = 0; j < K_J; ++j) {
            float w = Jreg[(size_t)j * V_N + v];
            atomicAdd(&jacc[j * 3 + 0], w * vs[0]);
            atomicAdd(&jacc[j * 3 + 1], w * vs[1]);
            atomicAdd(&jacc[j * 3 + 2], w * vs[2]);
        }
    }
    __syncthreads();
    if (t < 72) atomicAdd(&ws[OFF_J + t], jacc[t]);
}

// ---------------------------------------------------------------- kernel 2
// Rodrigues, lrotmin, kinematic chain, results[24][4][4] (single block)
__global__ void k_chain(const float* __restrict__ pose,
                        float* __restrict__ ws) {
    __shared__ float sR[K_J][9];
    int t = threadIdx.x;

    if (t < K_J) {
        float r0 = pose[t * 3 + 0], r1 = pose[t * 3 + 1], r2 = pose[t * 3 + 2];
        float a0 = r0 + 1e-8f, a1 = r1 + 1e-8f, a2 = r2 + 1e-8f;
        float theta = sqrtf(a0 * a0 + a1 * a1 + a2 * a2);
        float x = r0 / theta, y = r1 / theta, z = r2 / theta;
        float c = cosf(theta), s = sinf(theta), o = 1.0f - c;
        sR[t][0] = c + o * x * x;     sR[t][1] = o * x * y - s * z; sR[t][2] = o * x * z + s * y;
        sR[t][3] = o * y * x + s * z; sR[t][4] = c + o * y * y;     sR[t][5] = o * y * z - s * x;
        sR[t][6] = o * z * x - s * y; sR[t][7] = o * z * y + s * x; sR[t][8] = c + o * z * z;
    }
    __syncthreads();

    // lrotmin = (R[1:] - I).flatten
    if (t < (K_J - 1) * 9) {
        int j = t / 9 + 1, k = t - (j - 1) * 9;
        float e = sR[j][k] - ((k == 0 || k == 4 || k == 8) ? 1.0f : 0.0f);
        ws[OFF_LROT + t] = e;
    }

    if (t == 0) {
        const int par[K_J] = {0,0,0,0,1,2,3,4,5,6,7,8,9,9,9,12,13,14,16,17,18,19,20,21};
        float Jt[K_J][3];
        for (int j = 0; j < K_J; ++j)
            for (int c = 0; c < 3; ++c) Jt[j][c] = ws[OFF_J + j * 3 + c];

        float G[K_J][16];
        // G[0] = A(R0, J0)
        for (int a = 0; a < 3; ++a) {
            for (int b = 0; b < 3; ++b) G[0][a * 4 + b] = sR[0][a * 3 + b];
            G[0][a * 4 + 3] = Jt[0][a];
        }
        G[0][12] = 0.f; G[0][13] = 0.f; G[0][14] = 0.f; G[0][15] = 1.f;

        for (int i = 1; i < K_J; ++i) {
            int p = par[i];
            float A[16];
            for (int a = 0; a < 3; ++a) {
                for (int b = 0; b < 3; ++b) A[a * 4 + b] = sR[i][a * 3 + b];
                A[a * 4 + 3] = Jt[i][a] - Jt[p][a];
            }
            A[12] = 0.f; A[13] = 0.f; A[14] = 0.f; A[15] = 1.f;
            for (int a = 0; a < 4; ++a)
                for (int b = 0; b < 4; ++b) {
                    float acc = 0.f;
                    for (int k = 0; k < 4; ++k) acc += G[p][a * 4 + k] * A[k * 4 + b];
                    G[i][a * 4 + b] = acc;
                }
        }
        // results = G - pack(G @ [J;0])
        for (int j = 0; j < K_J; ++j) {
            for (int a = 0; a < 4; ++a) {
                float corr = G[j][a * 4 + 0] * Jt[j][0] +
                             G[j][a * 4 + 1] * Jt[j][1] +
                             G[j][a * 4 + 2] * Jt[j][2];
                ws[OFF_RES + j * 16 + a * 4 + 0] = G[j][a * 4 + 0];
                ws[OFF_RES + j * 16 + a * 4 + 1] = G[j][a * 4 + 1];
                ws[OFF_RES + j * 16 + a * 4 + 2] = G[j][a * 4 + 2];
                ws[OFF_RES + j * 16 + a * 4 + 3] = G[j][a * 4 + 3] - corr;
            }
        }
    }
}

// ---------------------------------------------------------------- kernel 3
// per 16-vertex wave tile:
//   A) v_posed = v_shaped + posedirs @ lrotmin   (wave-cooperative, coalesced)
//   B) T = weights[16x24] @ results[24x16]       (6x v_wmma_f32_16x16x4_f32)
//   C) out = T @ [v_posed,1] + trans
__global__ void __launch_bounds__(256)
k_final(const float* __restrict__ posedirs,
        const float* __restrict__ weights,
        const float* __restrict__ trans,
        const float* __restrict__ ws,
        float* __restrict__ out) {
    __shared__ float slrot[P_ + 1];
    __shared__ float sres[K_J * 16];
    __shared__ float svp[8][16][4];
    __shared__ float sT[8][16][16];

    int t = threadIdx.x;
    int lane = t & 31, wave = t >> 5;
    for (int i = t; i < P_;       i += 256) slrot[i] = ws[OFF_LROT + i];
    for (int i = t; i < K_J * 16; i += 256) sres[i]  = ws[OFF_RES + i];
    __syncthreads();

    int vbase = blockIdx.x * 128 + wave * 16;

    // --- phase A: wave-cooperative row dot-products, fully coalesced ---
    // Rows of this wave's 16 vertices are contiguous in posedirs:
    // global rows [vbase*3, vbase*3+48). Clamp the whole range for the tail.
    {
        int rbase = vbase * 3;
        if (rbase > (V_N - 16) * 3) rbase = (V_N - 16) * 3;   // clamped, uniform
        float lr0 = (lane < P_ - 192) ? slrot[192 + lane] : 0.0f; // tail weight
        for (int lr = 0; lr < 48; ++lr) {
            const float* pd = posedirs + (size_t)(rbase + lr) * P_;
            // prefetch next row's cache lines (7 x 128B cover 828B)
            if (lane < 7) __builtin_prefetch(pd + P_ + lane * 32, 0, 1);
            float partial;
            {   // 6 coalesced b32 loads + 1 tail (lanes 0..14)
                int p = lane;
                float s0 = pd[p] * slrot[p];             p += 32;
                float s1 = pd[p] * slrot[p];             p += 32;
                float s2 = pd[p] * slrot[p];             p += 32;
                float s3 = pd[p] * slrot[p];             p += 32;
                float s4 = pd[p] * slrot[p];             p += 32;
                float s5 = pd[p] * slrot[p];             p += 32;
                float s6 = (p < P_) ? pd[p] * lr0 : 0.0f;
                partial = ((s0 + s1) + (s2 + s3)) + ((s4 + s5) + s6);
            }
            #pragma unroll
            for (int off = 16; off > 0; off >>= 1)
                partial += __shfl_xor(partial, off, 32);
            if (lane == 0) {
                int vm = lr / 3, c = lr - vm * 3;
                svp[wave][vm][c] = partial + ws[OFF_VSH + rbase + lr];
            }
        }
    }
    if (lane < 16) svp[wave][lane][3] = 1.0f;
    __syncthreads();

    // --- phase B: T tile via f32 WMMA, EXEC all-ones (branchless clamp) ---
    int m    = lane & 15;
    int koff = (lane >> 4) * 2;                 // lanes 16-31 hold K+2,K+3
    int vcl  = vbase + m; if (vcl >= V_N) vcl = V_N - 1;
    v8f acc = {};
    #pragma unroll
    for (int k0 = 0; k0 < K_J; k0 += 4) {
        v2f A, B;
        A.x = weights[(size_t)vcl * K_J + k0 + koff];
        A.y = weights[(size_t)vcl * K_J + k0 + koff + 1];
        B.x = sres[(k0 + koff)     * 16 + m];
        B.y = sres[(k0 + koff + 1) * 16 + m];
        acc = __builtin_amdgcn_wmma_f32_16x16x4_f32(
                  false, A, false, B, (short)0, acc, false, false);
    }
    // scatter D (VGPR i: lanes0-15 -> M=i, lanes16-31 -> M=8+i)
    int mb = (lane >> 4) * 8;
    int n  = lane & 15;
    #pragma unroll
    for (int i = 0; i < 8; ++i) sT[wave][mb + i][n] = acc[i];
    __syncthreads();

    // --- phase C: v = T @ [v_posed,1] + trans ---
    // NOTE: svp was computed for the clamped vertex window; for tail blocks the
    // window is [V_N-16, V_N), matching the guarded stores below.
    int vwin = vbase; if (vwin > V_N - 16) vwin = V_N - 16;
    for (int idx = lane; idx < 48; idx += 32) {
        int vm = idx / 3, c = idx - vm * 3;
        int vv = vbase + vm;
        if (vv < V_N) {
            // vertex vv lives at window slot (vv - vwin)
            int wm = vv - vwin;
            float s = trans[c];
            #pragma unroll
            for (int b = 0; b < 4; ++b)
                s += sT[wave][vm][c * 4 + b] * svp[wave][wm][b];
            out[vv * 3 + c] = s;
        }
    }
}

// ---------------------------------------------------------------- launch
extern "C" void kernel_launch(void* const* d_in, const int* in_sizes, int n_in,
                              void* d_out, int out_size, void* d_ws, size_t ws_size,
                              hipStream_t stream) {
    const float* betas      = (const float*)d_in[0];
    const float* pose       = (const float*)d_in[1];
    const float* trans      = (const float*)d_in[2];
    const float* shapedirs  = (const float*)d_in[3];
    const float* v_template = (const float*)d_in[4];
    const float* Jreg       = (const float*)d_in[5];
    const float* weights    = (const float*)d_in[6];
    const float* posedirs   = (const float*)d_in[7];
    float* ws  = (float*)d_ws;
    float* out = (float*)d_out;

    k_zeroJ<<<1, 128, 0, stream>>>(ws);
    k_vshaped_J<<<(V_N + 255) / 256, 256, 0, stream>>>(betas, shapedirs, v_template, Jreg, ws);
    k_chain<<<1, 256, 0, stream>>>(pose, ws);
    k_final<<<(V_N + 127) / 128, 256, 0, stream>>>(posedirs, weights, trans, ws, out);
}